// Ses2Seq_45861660787539
// MI455X (gfx1250) — compile-verified
//
#include <hip/hip_runtime.h>
#include <hip/hip_bf16.h>

// MI455X / gfx1250 implementation.
//
// Theta_{k+1} = A @ Theta_k + B*u, A = 8578x8578 fp32 (294 MB > 192MB L2 ->
// streams from HBM every step). Batched over 8 sequences => [8578x8578]x[8578x8]
// GEMM per step; 256 steps => ~75 GB HBM => ~3.2 ms at 23.3 TB/s. Memory bound
// (8 MACs / 4B), so exact fp32 WMMA (v_wmma_f32_16x16x4_f32) is the right unit.
//
// This round: software-pipelined K loop (register double buffering, 16 b64
// loads in flight per wave) + scalarized (readfirstlane) loop bounds so the
// compiler emits a clean SGPR loop with partial s_wait_loadcnt, not a
// lock-step load->wait0->wmma chain.

typedef __attribute__((ext_vector_type(2))) float v2f;
typedef __attribute__((ext_vector_type(8))) float v8f;

#define LATENT      8578
#define PADROWS     8592                   // 537 * 16
#define BUF_FLOATS  ((PADROWS / 2) * 32)   // 137472 floats per Theta buffer
#define NTILE       537                    // 16-row tiles of A
#define MATVEC_BLOCKS 269                  // 2 tiles per block
#define GRID_BLOCKS (MATVEC_BLOCKS + 1)    // +1 MLP block
#define NSTEP       256
#define TW          256                    // timesteps per batch

// Theta buffer layout: pair-interleaved so the WMMA B operand is one v2f load:
// buf[(i>>1)*32 + b*2 + (i&1)] = Theta[i][b]; cols 8..15 and rows >= LATENT are 0.
__device__ __forceinline__ int thIdx(int i, int b) {
    return ((i >> 1) << 5) + (b << 1) + (i & 1);
}

__global__ void init_theta_kernel(const float* __restrict__ theta, float* __restrict__ ws) {
    int j = blockIdx.x * blockDim.x + threadIdx.x;
    if (j >= BUF_FLOATS) return;
    int p = j >> 5;
    int within = j & 31;
    int b = within >> 1;
    int i = (p << 1) + (within & 1);
    float v = (b < 8 && i < LATENT) ? theta[i] : 0.0f;
    ws[j] = v;                   // Theta_0 in buffer 0
    ws[BUF_FLOATS + j] = 0.0f;   // buffer 1 (pad slots stay 0 forever)
}

// Load one "body" = 8 K-chunks (32 K values): 8 A v2f + 8 Theta v2f per lane.
#define LOAD_BODY(abuf, bbuf, base)                                   \
    _Pragma("unroll") for (int q = 0; q < 8; ++q) {                   \
        abuf[q] = *(const v2f*)(pA  + (base) + q * 4);                \
        bbuf[q] = *(const v2f*)(pTb + ((base) + q * 4) * 16);         \
    }

#define WMMA_BODY(abuf, bbuf)                                         \
    _Pragma("unroll") for (int q = 0; q < 8; ++q) {                   \
        c = __builtin_amdgcn_wmma_f32_16x16x4_f32(                    \
                false, abuf[q], false, bbuf[q], (short)0, c, false, false); \
    }

__global__ __launch_bounds__(256)
void step_kernel(const float* __restrict__ A, const float* __restrict__ Bvec,
                 const float* __restrict__ xs, const float* __restrict__ ts,
                 float* __restrict__ ws, float* __restrict__ out, int step) {
    __shared__ float red[8][32][8];    // split-K partials (8 KB)
    __shared__ float hA[8][64];
    __shared__ float hB[8][64];

    const float* __restrict__ src = ws + (size_t)(step & 1) * BUF_FLOATS;        // Theta_k
    float* __restrict__       dst = ws + (size_t)((step + 1) & 1) * BUF_FLOATS;  // Theta_{k+1}

    const int tid  = threadIdx.x;
    const int lane = tid & 31;
    const int w    = tid >> 5;
    // Wave-uniform scalars -> SGPRs so loops/branches are scalar, EXEC untouched.
    const int wS   = __builtin_amdgcn_readfirstlane(w);

    if (blockIdx.x < MATVEC_BLOCKS) {
        if (step >= NSTEP) return;                 // final launch: MLP only
        const int tile = blockIdx.x * 2 + (wS >> 2);
        const int ks   = wS & 3;                   // split-K id (scalar)
        const int r0   = tile * 16;

        v8f c;
        #pragma unroll
        for (int r = 0; r < 8; ++r) c[r] = 0.0f;

        if (r0 < LATENT) {                         // scalar branch
            // A-operand layout: lane m holds row r0+m; lanes 16-31 at K+2.
            const int rowA = min(r0 + (lane & 15), LATENT - 1);
            const int koff = (lane >> 4) << 1;     // 0 or 2
            const float* pA  = A + (size_t)rowA * LATENT + koff;
            const float* pTb = src + ((lane & 15) << 1) + (koff << 4);

            // 2144 full chunks / 4 splits = 536 chunks = 67 bodies of 8.
            // Double-buffered pipeline: body j+1 loads issued before body j is consumed.
            v2f aA[8], bA[8], aB[8], bB[8];
            int base = ks * 536 * 4;               // float offset of this split's K range
            LOAD_BODY(aA, bA, base);               // body 0
            #pragma unroll 1
            for (int j = 0; j < 33; ++j) {
                LOAD_BODY(aB, bB, base + 32);      // body 2j+1
                __builtin_prefetch(pA + base + 256, 0, 0);   // A, ~8 bodies ahead (speculative)
                WMMA_BODY(aA, bA);                 // body 2j
                LOAD_BODY(aA, bA, base + 64);      // body 2j+2 (max index 66, always valid)
                WMMA_BODY(aB, bB);                 // body 2j+1
                base += 64;
            }
            WMMA_BODY(aA, bA);                     // body 66

            if (ks == 3) {                         // tail chunk: k0=8576, cols 8578/8579 invalid
                const int k0 = 8576;
                v2f a;
                if (lane < 16) a = *(const v2f*)(pA + k0);
                else { a[0] = 0.0f; a[1] = 0.0f; }
                v2f bm = *(const v2f*)(pTb + k0 * 16);       // padded Theta rows are 0
                c = __builtin_amdgcn_wmma_f32_16x16x4_f32(
                        false, a, false, bm, (short)0, c, false, false);
            }
        }

        #pragma unroll
        for (int r = 0; r < 8; ++r) red[w][lane][r] = c[r];
        __syncthreads();

        if ((wS & 3) == 0) {                       // one reducer wave per tile
            const int b  = lane & 15;              // D column = batch
            const int bb = b & 7;
            float u = 0.0f;
            if (step > 0) {
                float tc  = ts[bb * TW + step];
                float tp  = ts[bb * TW + step - 1];
                float xt  = xs[bb * TW + step];
                float xpp = (step == 1) ? xs[bb * TW]
                                        : out[bb * 2 * TW + (step - 2) * 2]; // ys[n-2][0]
                u = (xt - xpp) / (tc - tp + 1e-6f);
            }
            #pragma unroll
            for (int r = 0; r < 8; ++r) {
                float s = red[w][lane][r] + red[w + 1][lane][r]
                        + red[w + 2][lane][r] + red[w + 3][lane][r];
                const int i = r0 + r + ((lane >> 4) << 3);   // D: lanes>=16 hold M=r+8
                if (i < LATENT && b < 8) {
                    dst[thIdx(i, b)] = s + Bvec[i] * u;
                }
            }
        }
    } else {
        // ----- MLP block: ys[step-1] = MLP(Theta_step, t_{step-1}), wave w = batch -----
        if (step == 0) return;
        const int b = wS;
        const int n = step - 1;
        const float t = ts[b * TW + n];

        for (int i = lane; i < 64; i += 32) {      // h0 = relu(P0*t + P1)
            float v = src[thIdx(i, b)] * t + src[thIdx(64 + i, b)];
            hA[b][i] = fmaxf(v, 0.0f);
        }
        __syncthreads();
        for (int i = lane; i < 64; i += 32) {      // h1 = relu(P2@h0 + P3)
            float s = src[thIdx(4224 + i, b)];
            for (int j = 0; j < 64; ++j) s += src[thIdx(128 + i * 64 + j, b)] * hA[b][j];
            hB[b][i] = fmaxf(s, 0.0f);
        }
        __syncthreads();
        for (int i = lane; i < 64; i += 32) {      // h2 = relu(P4@h1 + P5)
            float s = src[thIdx(8384 + i, b)];
            for (int j = 0; j < 64; ++j) s += src[thIdx(4288 + i * 64 + j, b)] * hB[b][j];
            hA[b][i] = fmaxf(s, 0.0f);
        }
        __syncthreads();
        if (lane < 2) {                            // out = P6@h2 + P7
            float s = src[thIdx(8576 + lane, b)];
            for (int j = 0; j < 64; ++j) s += src[thIdx(8448 + lane * 64 + j, b)] * hA[b][j];
            out[b * 2 * TW + n * 2 + lane] = s;
        }
    }
}

extern "C" void kernel_launch(void* const* d_in, const int* in_sizes, int n_in,
                              void* d_out, int out_size, void* d_ws, size_t ws_size,
                              hipStream_t stream) {
    const float* xs    = (const float*)d_in[0];   // (8,256,1)
    const float* ts    = (const float*)d_in[1];   // (8,256)
    const float* A     = (const float*)d_in[2];   // (8578,8578)
    const float* Bvec  = (const float*)d_in[3];   // (8578,1)
    const float* theta = (const float*)d_in[4];   // (8578,)
    float* out = (float*)d_out;                   // (8,256,2)
    float* ws  = (float*)d_ws;                    // needs 2*BUF_FLOATS*4 ~= 1.1 MB

    init_theta_kernel<<<(BUF_FLOATS + 255) / 256, 256, 0, stream>>>(theta, ws);
    for (int k = 0; k <= NSTEP; ++k) {
        step_kernel<<<GRID_BLOCKS, 256, 0, stream>>>(A, Bvec, xs, ts, ws, out, k);
    }
}